// SSDClassScoresExtractor_17678085390810
// MI455X (gfx1250) — compile-verified
//
#include <hip/hip_runtime.h>
#include <cstdint>
#include <cmath>

// ---------------- problem constants ----------------
constexpr int NB = 16;          // batch
constexpr int NA = 8732;        // anchors
constexpr int NC = 91;          // classes (incl background)
constexpr int NFG = 90;         // foreground classes
constexpr int KTOP = 400;       // per-class top-k
constexpr int ND = 200;         // detections kept
constexpr int NCAND = NFG * KTOP;        // 36000 candidates / image
constexpr float IMG = 300.0f;
constexpr float XFORM_CLIP = 4.135166556742356f;  // log(1000/16)
constexpr float SCORE_T = 0.01f;
constexpr float NMS_T = 0.45f;
constexpr unsigned KEY_NEGINF = 0x007FFFFFu;      // keyOfFloat(-inf)

// ---------------- helpers ----------------
__device__ __forceinline__ uint32_t ldsAddr(const void* p) {
  // generic pointer to LDS: low 32 bits are the LDS byte address (ISA 10.2 aperture rule)
  return (uint32_t)(uintptr_t)p;
}
// monotonic float -> uint key (larger key == larger float, NaN-free domain)
__device__ __forceinline__ unsigned keyOfBits(unsigned b) {
  return (b & 0x80000000u) ? ~b : (b | 0x80000000u);
}
__device__ __forceinline__ unsigned keyOfFloat(float f) { return keyOfBits(__float_as_uint(f)); }
__device__ __forceinline__ float floatOfKey(unsigned k) {
  return __uint_as_float((k & 0x80000000u) ? (k ^ 0x80000000u) : ~k);
}

// async global->LDS staging primitives (gfx1250)
__device__ __forceinline__ void asyncLdsB32(uint32_t lds, uint32_t voff, const void* sbase) {
  asm volatile("global_load_async_to_lds_b32 %0, %1, %2 offset:0"
               :: "v"(lds), "v"(voff), "s"(sbase) : "memory");
}
__device__ __forceinline__ void asyncLdsB128(uint32_t lds, uint32_t voff, const void* sbase) {
  asm volatile("global_load_async_to_lds_b128 %0, %1, %2 offset:0"
               :: "v"(lds), "v"(voff), "s"(sbase) : "memory");
}
__device__ __forceinline__ void waitAsync0() {
  asm volatile("s_wait_asynccnt 0x0" ::: "memory");
}

// ---------------- K1: softmax stats + masked transposed scores ----------------
__global__ __launch_bounds__(256) void k_softmax_stats(
    const float* __restrict__ logits, float* __restrict__ scoresT, float* __restrict__ stats) {
  constexpr int CH = 64;                // anchors per block
  constexpr int ELEMS = CH * NC;        // 5824 floats
  __shared__ float lg[ELEMS];           // staged logits [anchor][class]
  __shared__ float ms[NFG * CH];        // masked scores transposed [fgclass][anchor]
  const int t = threadIdx.x;
  const int b = blockIdx.y;
  const int a0 = blockIdx.x * CH;
  const int nAv = min(CH, NA - a0);
  const size_t baseElem = ((size_t)b * NA + a0) * NC;
  const int totRem = (int)((size_t)NB * NA * NC - baseElem);
  const float* gsrc = logits + baseElem;
  const uint32_t lb = ldsAddr(lg);
  // async stage (b32: row start only 4B-aligned). Clamp OOB tail to last valid element.
  for (int k = 0; k < 23; ++k) {
    int i = t + k * 256;
    if (i < ELEMS) {
      int gi = (i < totRem) ? i : (totRem - 1);
      asyncLdsB32(lb + (uint32_t)i * 4u, (uint32_t)gi * 4u, gsrc);
    }
  }
  waitAsync0();
  __syncthreads();
  const int w = t >> 5, l = t & 31;
  for (int r = 0; r < 8; ++r) {          // 8 anchors per wave
    const int la = w * 8 + r;
    if (la >= nAv) break;                // uniform per wave
    const float* rowp = &lg[la * NC];
    float v0 = rowp[l];
    float v1 = rowp[l + 32];
    float v2 = (l < NC - 64) ? rowp[l + 64] : -INFINITY;
    float mx = fmaxf(fmaxf(v0, v1), v2);
    for (int m = 16; m; m >>= 1) mx = fmaxf(mx, __shfl_xor(mx, m, 32));
    float e0 = __expf(v0 - mx), e1 = __expf(v1 - mx);
    float e2 = (l < NC - 64) ? __expf(v2 - mx) : 0.f;
    float sum = e0 + e1 + e2;
    for (int m = 16; m; m >>= 1) sum += __shfl_xor(sum, m, 32);
    float inv = 1.f / sum;
    float p0 = e0 * inv, p1 = e1 * inv, p2 = e2 * inv;
    if (l >= 1) ms[(l - 1) * CH + la] = (p0 > SCORE_T) ? p0 : -1.f;   // class l     -> fg l-1
    ms[(l + 31) * CH + la] = (p1 > SCORE_T) ? p1 : -1.f;              // class l+32  -> fg l+31
    if (l < NC - 64) ms[(l + 63) * CH + la] = (p2 > SCORE_T) ? p2 : -1.f; // class l+64 -> fg l+63
    if (l == 0) {
      size_t si = ((size_t)b * NA + a0 + la) * 2;
      stats[si] = mx; stats[si + 1] = sum;
    }
  }
  __syncthreads();
  // coalesced transposed write: [b][fgclass][anchor]
  for (int i = t; i < NFG * CH; i += 256) {
    int c = i >> 6, la = i & 63;
    if (la < nAv) scoresT[((size_t)b * NFG + c) * NA + a0 + la] = ms[i];
  }
}

// ---------------- K2: box decode ----------------
__global__ __launch_bounds__(256) void k_decode(
    const float* __restrict__ rel, const float* __restrict__ anch, float* __restrict__ boxes) {
  int idx = blockIdx.x * 256 + threadIdx.x;
  if (idx >= NB * NA) return;
  int a = idx % NA;
  float4 r = ((const float4*)rel)[idx];
  float4 an = ((const float4*)anch)[a];
  float wa = an.z - an.x, ha = an.w - an.y;
  float cxa = an.x + 0.5f * wa, cya = an.y + 0.5f * ha;
  float dx = r.x * 0.1f, dy = r.y * 0.1f;
  float dw = fminf(r.z * 0.2f, XFORM_CLIP);
  float dh = fminf(r.w * 0.2f, XFORM_CLIP);
  float cx = dx * wa + cxa, cy = dy * ha + cya;
  float w = __expf(dw) * wa, h = __expf(dh) * ha;
  float4 o;
  o.x = fminf(fmaxf(cx - 0.5f * w, 0.f), IMG);
  o.y = fminf(fmaxf(cy - 0.5f * h, 0.f), IMG);
  o.z = fminf(fmaxf(cx + 0.5f * w, 0.f), IMG);
  o.w = fminf(fmaxf(cy + 0.5f * h, 0.f), IMG);
  ((float4*)boxes)[idx] = o;
}

// ---------------- shared radix select (descending top-KSEL threshold) ----------------
// out: bc[0]=threshold key T, bc[1]=count(keys>T), bc[2]=needed count of ==T
__device__ void radixSelectTopK(const unsigned* keys, int N, int KSEL,
                                unsigned* hist, unsigned* bc) {
  const int t = threadIdx.x;
  unsigned prefix = 0;
  int kneed = KSEL, cntGt = 0;
  for (int pass = 0; pass < 4; ++pass) {
    const int shift = 24 - 8 * pass;
    const unsigned pmask = (pass == 0) ? 0u : (0xFFFFFFFFu << (shift + 8));
    if (t < 256) hist[t] = 0;
    __syncthreads();
    for (int i = t; i < N; i += 256) {
      unsigned k = keys[i];
      if ((k & pmask) == prefix) atomicAdd(&hist[(k >> shift) & 255u], 1u);  // LDS atomic, order-independent
    }
    __syncthreads();
    if (t == 0) {
      int acc = 0; unsigned dsel = 0;
      for (int d = 255; d >= 0; --d) {
        int cnt = (int)hist[d];
        if (acc + cnt >= kneed) { dsel = (unsigned)d; break; }
        acc += cnt;
      }
      bc[0] = dsel; bc[1] = (unsigned)acc;
    }
    __syncthreads();
    prefix |= bc[0] << shift;
    cntGt += (int)bc[1];
    kneed -= (int)bc[1];
    __syncthreads();
  }
  if (t == 0) { bc[0] = prefix; bc[1] = (unsigned)cntGt; bc[2] = (unsigned)kneed; }
  __syncthreads();
}

// ---------------- K3: per (image,class) top-400 ----------------
__global__ __launch_bounds__(256) void k_topk(
    const float* __restrict__ scoresT, float* __restrict__ cand_score,
    int* __restrict__ cand_anchor) {
  extern __shared__ unsigned keys[];           // NA entries (34928 B dynamic)
  __shared__ unsigned hist[256];
  __shared__ int scanA[256], scanB[256];
  __shared__ unsigned bc[4];
  const int c = blockIdx.x, b = blockIdx.y, t = threadIdx.x;
  const float* row = scoresT + ((size_t)b * NFG + c) * NA;   // 16B-aligned row
  const uint32_t lb = ldsAddr(keys);
  for (int k = 0; k < 9; ++k) {
    int i4 = t + k * 256;
    if (i4 < NA / 4) asyncLdsB128(lb + (uint32_t)i4 * 16u, (uint32_t)i4 * 16u, row);
  }
  waitAsync0();
  __syncthreads();
  for (int i = t; i < NA; i += 256) keys[i] = keyOfBits(keys[i]);
  __syncthreads();
  radixSelectTopK(keys, NA, KTOP, hist, bc);
  const unsigned T = bc[0]; const int cntGt = (int)bc[1];
  // deterministic index-ordered compaction
  const int chunk = (NA + 255) / 256;
  int lo = t * chunk, hi = min(lo + chunk, NA);
  int cg = 0, ce = 0;
  for (int i = lo; i < hi; ++i) { unsigned k = keys[i]; cg += (k > T); ce += (k == T); }
  scanA[t] = cg; scanB[t] = ce;
  __syncthreads();
  if (t == 0) {
    int a = 0, e = 0;
    for (int i = 0; i < 256; ++i) { int g = scanA[i], q = scanB[i]; scanA[i] = a; scanB[i] = e; a += g; e += q; }
  }
  __syncthreads();
  int pg = scanA[t], pe = scanB[t];
  const size_t base = ((size_t)b * NFG + c) * KTOP;
  for (int i = lo; i < hi; ++i) {
    unsigned k = keys[i];
    if (k > T) { cand_score[base + pg] = floatOfKey(k); cand_anchor[base + pg] = i; ++pg; }
    else if (k == T) {
      int pos = cntGt + pe++;
      if (pos < KTOP) { cand_score[base + pos] = floatOfKey(k); cand_anchor[base + pos] = i; }
    }
  }
}

// ---------------- K4: per-class greedy NMS (classes independent: offsets kill cross-class IoU) ----------------
__global__ __launch_bounds__(256) void k_nms_class(
    const float* __restrict__ cand_score, const int* __restrict__ cand_anchor,
    const float* __restrict__ boxes, float* __restrict__ surv) {
  __shared__ float x1s[KTOP], y1s[KTOP], x2s[KTOP], y2s[KTOP], ss[KTOP], orig[KTOP];
  __shared__ int sflag[KTOP];
  __shared__ unsigned long long red[8];
  const int c = blockIdx.x, b = blockIdx.y, t = threadIdx.x;
  const size_t base = ((size_t)b * NFG + c) * KTOP;
  for (int i = t; i < KTOP; i += 256) {
    float sc = cand_score[base + i];
    int a = cand_anchor[base + i];
    const float* bp = boxes + ((size_t)b * NA + a) * 4;
    x1s[i] = bp[0]; y1s[i] = bp[1]; x2s[i] = bp[2]; y2s[i] = bp[3];
    orig[i] = sc;
    ss[i] = (sc > SCORE_T) ? sc : -INFINITY;
    sflag[i] = 0;
  }
  __syncthreads();
  for (int iter = 0; iter < KTOP; ++iter) {
    // argmax with lowest-index tie-break: key = (scoreKey<<32) | ~idx
    unsigned long long best = 0ull;
    for (int i = t; i < KTOP; i += 256) {
      unsigned long long key =
          ((unsigned long long)keyOfFloat(ss[i]) << 32) | (unsigned)~(unsigned)i;
      if (key > best) best = key;
    }
    for (int m = 16; m; m >>= 1) {
      unsigned long long o = __shfl_xor(best, m, 32);
      if (o > best) best = o;
    }
    if ((t & 31) == 0) red[t >> 5] = best;
    __syncthreads();
    unsigned long long mBest = red[0];
    for (int i = 1; i < 8; ++i) if (red[i] > mBest) mBest = red[i];
    unsigned bk = (unsigned)(mBest >> 32);
    if (bk <= KEY_NEGINF) break;           // uniform: everything suppressed
    int j = (int)~(unsigned)(mBest & 0xFFFFFFFFu);
    if (t == 0) sflag[j] = 1;
    float jx1 = x1s[j], jy1 = y1s[j], jx2 = x2s[j], jy2 = y2s[j];
    float ja = (jx2 - jx1) * (jy2 - jy1);
    for (int i = t; i < KTOP; i += 256) {
      float xx1 = fmaxf(jx1, x1s[i]), yy1 = fmaxf(jy1, y1s[i]);
      float xx2 = fminf(jx2, x2s[i]), yy2 = fminf(jy2, y2s[i]);
      float inter = fmaxf(xx2 - xx1, 0.f) * fmaxf(yy2 - yy1, 0.f);
      float a2 = (x2s[i] - x1s[i]) * (y2s[i] - y1s[i]);
      float iou = inter / (ja + a2 - inter + 1e-9f);
      if (iou > NMS_T) ss[i] = -INFINITY;  // includes i==j (self IoU = 1)
    }
    __syncthreads();
  }
  __syncthreads();
  for (int i = t; i < KTOP; i += 256)
    surv[(size_t)b * NCAND + c * KTOP + i] = sflag[i] ? orig[i] : -INFINITY;
}

// ---------------- K5: per-image top-200 survivors in selection order ----------------
__global__ __launch_bounds__(256) void k_top200(
    const float* __restrict__ surv, int* __restrict__ keep, float* __restrict__ kscore) {
  extern __shared__ unsigned keys[];          // NCAND entries (144000 B dynamic)
  __shared__ unsigned hist[256];
  __shared__ int scanA[256], scanB[256];
  __shared__ unsigned bc[4];
  __shared__ unsigned long long pairs[256];
  const int t = threadIdx.x, b = blockIdx.x;
  const float* row = surv + (size_t)b * NCAND;
  const uint32_t lb = ldsAddr(keys);
  for (int k = 0; k < 36; ++k) {
    int i4 = t + k * 256;
    if (i4 < NCAND / 4) asyncLdsB128(lb + (uint32_t)i4 * 16u, (uint32_t)i4 * 16u, row);
  }
  waitAsync0();
  __syncthreads();
  for (int i = t; i < NCAND; i += 256) keys[i] = keyOfBits(keys[i]);
  pairs[t] = 0ull;
  __syncthreads();
  radixSelectTopK(keys, NCAND, ND, hist, bc);
  const unsigned T = bc[0]; const int cntGt = (int)bc[1];
  const int chunk = (NCAND + 255) / 256;
  int lo = t * chunk, hi = min(lo + chunk, NCAND);
  int cg = 0, ce = 0;
  for (int i = lo; i < hi; ++i) { unsigned k = keys[i]; cg += (k > T); ce += (k == T); }
  scanA[t] = cg; scanB[t] = ce;
  __syncthreads();
  if (t == 0) {
    int a = 0, e = 0;
    for (int i = 0; i < 256; ++i) { int g = scanA[i], q = scanB[i]; scanA[i] = a; scanB[i] = e; a += g; e += q; }
  }
  __syncthreads();
  int pg = scanA[t], pe = scanB[t];
  for (int i = lo; i < hi; ++i) {
    unsigned k = keys[i];
    unsigned long long pr = ((unsigned long long)k << 32) | (unsigned)~(unsigned)i;
    if (k > T) pairs[pg++] = pr;
    else if (k == T) { int pos = cntGt + pe++; if (pos < ND) pairs[pos] = pr; }
  }
  __syncthreads();
  // bitonic sort descending (score desc, index asc) over 256 padded pairs
  for (int size = 2; size <= 256; size <<= 1) {
    for (int stride = size >> 1; stride > 0; stride >>= 1) {
      __syncthreads();
      int j = t ^ stride;
      if (j > t) {
        bool up = ((t & size) == 0);
        unsigned long long x = pairs[t], y = pairs[j];
        bool sw = up ? (x < y) : (x > y);
        if (sw) { pairs[t] = y; pairs[j] = x; }
      }
    }
  }
  __syncthreads();
  if (t < ND) {
    unsigned long long pr = pairs[t];
    unsigned k = (unsigned)(pr >> 32);
    int idx = (int)~(unsigned)(pr & 0xFFFFFFFFu);
    bool valid = (k > KEY_NEGINF);           // only real (positive) survivor scores
    keep[b * ND + t] = valid ? idx : 0;
    kscore[b * ND + t] = valid ? floatOfKey(k) : -INFINITY;
  }
}

// ---------------- K6: finalize outputs (one wave per detection) ----------------
__global__ __launch_bounds__(128) void k_finalize(
    const float* __restrict__ logits, const float* __restrict__ stats,
    const float* __restrict__ boxes, const int* __restrict__ cand_anchor,
    const int* __restrict__ keep, const float* __restrict__ kscore,
    float* __restrict__ outBoxes, float* __restrict__ outScores, float* __restrict__ outLabels) {
  const int b = blockIdx.y;
  const int det = blockIdx.x * 4 + (threadIdx.x >> 5);
  const int l = threadIdx.x & 31;
  if (det >= ND) return;
  float ks = kscore[b * ND + det];
  bool valid = (ks > -INFINITY);
  int n = keep[b * ND + det];
  int anchor = valid ? cand_anchor[(size_t)b * NCAND + n] : 0;
  const size_t bo = (size_t)b * ND + det;
  if (l < 4) outBoxes[bo * 4 + l] = valid ? boxes[((size_t)b * NA + anchor) * 4 + l] : 0.f;
  float mx = stats[((size_t)b * NA + anchor) * 2];
  float inv = 1.f / stats[((size_t)b * NA + anchor) * 2 + 1];
  const float* lrow = logits + ((size_t)b * NA + anchor) * NC;
  float b1 = -INFINITY, b2 = -INFINITY; int i1 = NC, i2 = NC;
  for (int c = l; c < NC; c += 32) {
    if (c == 0) continue;                    // foreground only
    float p = __expf(lrow[c] - mx) * inv;
    if (p > b1) { b2 = b1; i2 = i1; b1 = p; i1 = c; }
    else if (p > b2) { b2 = p; i2 = c; }
  }
  for (int m = 16; m; m >>= 1) {
    float o1 = __shfl_xor(b1, m, 32), o2 = __shfl_xor(b2, m, 32);
    int oi1 = __shfl_xor(i1, m, 32), oi2 = __shfl_xor(i2, m, 32);
    float n1, n2; int ni1, ni2;
    bool aFirst = (b1 > o1) || (b1 == o1 && i1 < oi1);
    if (aFirst) {
      n1 = b1; ni1 = i1;
      bool sec = (o1 > b2) || (o1 == b2 && oi1 < i2);
      if (sec) { n2 = o1; ni2 = oi1; } else { n2 = b2; ni2 = i2; }
    } else {
      n1 = o1; ni1 = oi1;
      bool sec = (b1 > o2) || (b1 == o2 && i1 < oi2);
      if (sec) { n2 = b1; ni2 = i1; } else { n2 = o2; ni2 = oi2; }
    }
    b1 = n1; i1 = ni1; b2 = n2; i2 = ni2;
  }
  if (l == 0) {
    outScores[bo * 2]     = valid ? b1 : 0.f;
    outScores[bo * 2 + 1] = valid ? b2 : 0.f;
    outLabels[bo * 2]     = valid ? (float)i1 : 0.f;   // class index == tidx+1
    outLabels[bo * 2 + 1] = valid ? (float)i2 : 0.f;
  }
}

// ---------------- launch ----------------
extern "C" void kernel_launch(void* const* d_in, const int* in_sizes, int n_in,
                              void* d_out, int out_size, void* d_ws, size_t ws_size,
                              hipStream_t stream) {
  (void)in_sizes; (void)n_in; (void)out_size; (void)ws_size;
  const float* logits = (const float*)d_in[0];  // [B,A,C]
  const float* rel    = (const float*)d_in[1];  // [B,A,4]
  const float* anch   = (const float*)d_in[2];  // [A,4]

  char* ws = (char*)d_ws;
  size_t off = 0;
  auto take = [&](size_t bytes) -> void* {
    void* p = ws + off;
    off += (bytes + 255) & ~(size_t)255;
    return p;
  };
  float* stats       = (float*)take((size_t)NB * NA * 2 * sizeof(float));       // ~1.1 MB
  float* scoresT     = (float*)take((size_t)NB * NFG * NA * sizeof(float));     // ~50.3 MB
  float* boxes       = (float*)take((size_t)NB * NA * 4 * sizeof(float));       // ~2.2 MB
  float* cand_score  = (float*)take((size_t)NB * NCAND * sizeof(float));        // ~2.3 MB
  int*   cand_anchor = (int*)  take((size_t)NB * NCAND * sizeof(int));          // ~2.3 MB
  float* surv        = (float*)take((size_t)NB * NCAND * sizeof(float));        // ~2.3 MB
  int*   keep        = (int*)  take((size_t)NB * ND * sizeof(int));
  float* kscore      = (float*)take((size_t)NB * ND * sizeof(float));

  float* outBoxes  = (float*)d_out;                          // [B,D,4]
  float* outScores = outBoxes + (size_t)NB * ND * 4;         // [B,D,2]
  float* outLabels = outScores + (size_t)NB * ND * 2;        // [B,D,2]

  dim3 g1((NA + 63) / 64, NB);
  k_softmax_stats<<<g1, 256, 0, stream>>>(logits, scoresT, stats);
  k_decode<<<(NB * NA + 255) / 256, 256, 0, stream>>>(rel, anch, boxes);
  dim3 g2(NFG, NB);
  k_topk<<<g2, 256, (size_t)NA * sizeof(unsigned), stream>>>(scoresT, cand_score, cand_anchor);
  k_nms_class<<<g2, 256, 0, stream>>>(cand_score, cand_anchor, boxes, surv);
  k_top200<<<NB, 256, (size_t)NCAND * sizeof(unsigned), stream>>>(surv, keep, kscore);
  dim3 g6((ND + 3) / 4, NB);
  k_finalize<<<g6, 128, 0, stream>>>(logits, stats, boxes, cand_anchor, keep, kscore,
                                     outBoxes, outScores, outLabels);
}